// TradingSNN_52776558133341
// MI455X (gfx1250) — compile-verified
//
#include <hip/hip_runtime.h>

// ---------------------------------------------------------------------------
// Fused SNN forward for MI455X (gfx1250, wave32).
//   cur1 = x@W1^T + b1  (K=5, computed once, kept in VGPRs)
//   20x: mem1 = B*mem1 + cur1 - spk1_prev ; spk1 = mem1>1
//        cur2 = spk1 @ W2^T + b2  -> done as WMMA: D = W2_pad(16x100) x spk1^T(100x16)
//        mem2/spk2/cnt analogous (3 outputs)
// One wave owns 16 batch rows. State lives in registers in WMMA operand
// layout: lane l holds batch row (l&15), K indices 4c + 2*(l>>4) + {0,1}.
// Matrix op: v_wmma_f32_16x16x4_f32, 25 chained per step (K=100 = 25*4 exact,
// full f32 precision), co-executing with the VALU recurrence.
// ---------------------------------------------------------------------------

typedef float v2f __attribute__((ext_vector_type(2)));
typedef float v8f __attribute__((ext_vector_type(8)));

#define BETA  0.95f
#define NI    5
#define NH    100
#define NO    3
#define NCH   25            // NH / 4 K-chunks for 16x16x4 WMMA
#define ROWS_PER_WAVE 16
#define WAVES_PER_BLOCK 8
#define ROWS_PER_BLOCK (ROWS_PER_WAVE * WAVES_PER_BLOCK)

__global__ __launch_bounds__(256)
void TradingSNN_snn_wmma_kernel(const float* __restrict__ x,
                                const float* __restrict__ W1,
                                const float* __restrict__ b1,
                                const float* __restrict__ W2,
                                const float* __restrict__ b2,
                                const int*   __restrict__ tsteps,
                                float* __restrict__ out,
                                int B)
{
    __shared__ float sW1[NH * NI];
    __shared__ float sb1[NH];
    __shared__ float sW2[NO * NH];

    // cooperative stage of weights into LDS (one-time, ~3.6KB)
    for (int i = threadIdx.x; i < NH * NI; i += blockDim.x) sW1[i] = W1[i];
    for (int i = threadIdx.x; i < NH;      i += blockDim.x) sb1[i] = b1[i];
    for (int i = threadIdx.x; i < NO * NH; i += blockDim.x) sW2[i] = W2[i];
    __syncthreads();

    const int lane = threadIdx.x & 31;        // wave32
    const int wave = threadIdx.x >> 5;
    const int n    = lane & 15;               // batch row within tile (B-operand col)
                                              // == output-neuron row for A operand
    const int half = lane >> 4;               // K-half: lanes<16 -> K%4 in {0,1}
    long row = (long)blockIdx.x * ROWS_PER_BLOCK + wave * ROWS_PER_WAVE + n;
    const bool valid = row < B;
    if (row >= B) row = B - 1;                // clamp loads; EXEC stays full for WMMA

    // per-lane x values for this batch row (5 floats, 20B)
    float xv[NI];
#pragma unroll
    for (int i = 0; i < NI; ++i) xv[i] = x[row * NI + i];

    // Recurrent state in WMMA B-operand layout: value j = 2*c + v maps to
    // hidden unit h = 4*c + 2*half + v for this lane's batch row.
    float cur1[2 * NCH];
    float mem1[2 * NCH];
    float spk1[2 * NCH];
    float w2a [2 * NCH];   // W2 zero-padded to 16 rows, in A-operand layout (constant)

#pragma unroll
    for (int c = 0; c < NCH; ++c) {
#pragma unroll
        for (int v = 0; v < 2; ++v) {
            const int h = 4 * c + 2 * half + v;
            float a = sb1[h];
#pragma unroll
            for (int i = 0; i < NI; ++i) a += xv[i] * sW1[h * NI + i];
            cur1[2 * c + v] = a;
            mem1[2 * c + v] = 0.0f;
            spk1[2 * c + v] = 0.0f;                       // reset at t=0 is 0
            w2a [2 * c + v] = (n < NO) ? sW2[n * NH + h] : 0.0f;
        }
    }

    float b2v[NO];
#pragma unroll
    for (int o = 0; o < NO; ++o) b2v[o] = b2[o];          // uniform -> scalar loads

    float mem2[NO] = {0.f, 0.f, 0.f};
    float spk2[NO] = {0.f, 0.f, 0.f};
    float cnt [NO] = {0.f, 0.f, 0.f};

    const int T = tsteps[0];
    for (int t = 0; t < T; ++t) {
        // Layer-1 LIF update: reset == previous spike (stop-gradient identical
        // in forward). 4 VALU ops per value, 50 values per lane.
#pragma unroll
        for (int j = 0; j < 2 * NCH; ++j) {
            mem1[j] = BETA * mem1[j] + (cur1[j] - spk1[j]);
            spk1[j] = (mem1[j] > 1.0f) ? 1.0f : 0.0f;
        }

        // cur2^T = W2_pad(16x100) x spk1^T(100x16) via 25 chained f32 WMMAs.
        v8f acc = {0.f, 0.f, 0.f, 0.f, 0.f, 0.f, 0.f, 0.f};
#pragma unroll
        for (int c = 0; c < NCH; ++c) {
            v2f a = {w2a [2 * c], w2a [2 * c + 1]};
            v2f b = {spk1[2 * c], spk1[2 * c + 1]};
            acc = __builtin_amdgcn_wmma_f32_16x16x4_f32(
                    /*neg_a=*/false, a, /*neg_b=*/false, b,
                    /*c_mod=*/(short)0, acc, /*reuse_a=*/false, /*reuse_b=*/false);
        }

        // Layer-2 LIF + spike count. D-layout: lanes 0-15 VGPR o == output o,
        // batch col == lane. Lanes 16-31 hold zero-padded rows (never stored).
#pragma unroll
        for (int o = 0; o < NO; ++o) {
            const float cur2 = acc[o] + b2v[o];
            mem2[o] = BETA * mem2[o] + (cur2 - spk2[o]);
            spk2[o] = (mem2[o] > 1.0f) ? 1.0f : 0.0f;
            cnt[o] += spk2[o];
        }
    }

    if (valid && lane < 16) {
#pragma unroll
        for (int o = 0; o < NO; ++o) out[row * NO + o] = cnt[o];
    }
}

extern "C" void kernel_launch(void* const* d_in, const int* in_sizes, int n_in,
                              void* d_out, int out_size, void* d_ws, size_t ws_size,
                              hipStream_t stream) {
    const float* x  = (const float*)d_in[0];
    const float* W1 = (const float*)d_in[1];
    const float* b1 = (const float*)d_in[2];
    const float* W2 = (const float*)d_in[3];
    const float* b2 = (const float*)d_in[4];
    const int*   ts = (const int*)  d_in[5];
    float* out = (float*)d_out;

    const int B = in_sizes[0] / NI;                       // 262144
    const int grid = (B + ROWS_PER_BLOCK - 1) / ROWS_PER_BLOCK;  // 2048 blocks
    TradingSNN_snn_wmma_kernel<<<grid, WAVES_PER_BLOCK * 32, 0, stream>>>(
        x, W1, b1, W2, b2, ts, out, B);
}